// VocabProjector_6949257085491
// MI455X (gfx1250) — compile-verified
//
#include <hip/hip_runtime.h>
#include <stdint.h>
#include <stddef.h>

#ifndef __has_builtin
#define __has_builtin(x) 0
#endif

#if __has_builtin(__builtin_amdgcn_global_load_async_to_lds_b128) && \
    __has_builtin(__builtin_amdgcn_s_wait_asynccnt)
#define USE_ASYNC_LDS 1
#else
#define USE_ASYNC_LDS 0
#endif

#define TEACHER_VOCAB 128000
#define STUDENT_VOCAB 32000
#define TOPK   256
#define NROWS  2048                       /* B*T = 2*1024 */
#define WG     512                        /* 16 wave32s */
#define ROWF4  (TEACHER_VOCAB / 4)        /* 32000 float4 per row */
#define TILEF4 WG                         /* 512 float4 = 2048 floats per tile */
#define NTILES ((ROWF4 + TILEF4 - 1) / TILEF4)  /* 63 */
#define INV_T  0.25f                      /* 1 / TEMPERATURE */

typedef int v4i __attribute__((ext_vector_type(4)));
#define AS1 __attribute__((address_space(1)))
#define AS3 __attribute__((address_space(3)))

// Monotone float -> uint32 key (order-preserving for all finite floats).
__device__ __forceinline__ unsigned f2k(float f) {
  unsigned u = __float_as_uint(f);
  return u ^ ((unsigned)((int)u >> 31) | 0x80000000u);
}
__device__ __forceinline__ float k2f(unsigned k) {
  unsigned u = (k & 0x80000000u) ? (k ^ 0x80000000u) : ~k;
  return __uint_as_float(u);
}

// Async global -> LDS 16B copy (CDNA5 GLOBAL_LOAD_ASYNC_TO_LDS_B128, ASYNCcnt).
__device__ __forceinline__ void async_ld_b128(const float4* g, float4* l) {
#if USE_ASYNC_LDS
  __builtin_amdgcn_global_load_async_to_lds_b128(
      (AS1 v4i*)g, (AS3 v4i*)l, /*offset=*/0, /*cpol=*/0);
#else
  *l = *g;
#endif
}

template <int N>
__device__ __forceinline__ void wait_asynccnt() {
#if USE_ASYNC_LDS
  __builtin_amdgcn_s_wait_asynccnt(N);
#endif
}

// Stream one 128000-float row through LDS with double-buffered async copies.
// Calls fn(global_element_index, value) for every element exactly once.
template <typename F>
__device__ __forceinline__ void stream_row(const float4* __restrict__ row4,
                                           float4 (&buf)[2][TILEF4], F&& fn) {
  const int tid = threadIdx.x;
  // prime tile 0
  async_ld_b128(row4 + tid, &buf[0][tid]);
  for (int t = 0; t < NTILES; ++t) {
    if (t + 1 < NTILES) {
      int idx = (t + 1) * TILEF4 + tid;
      if (idx >= ROWF4) idx = ROWF4 - 1;    // clamp: keeps wait-count uniform
      async_ld_b128(row4 + idx, &buf[(t + 1) & 1][tid]);
      wait_asynccnt<1>();                   // tile t is done (in-order)
    } else {
      wait_asynccnt<0>();
    }
    __syncthreads();                        // all waves' slices of tile t landed
    int idx = t * TILEF4 + tid;
    if (idx < ROWF4) {
      float4 v = buf[t & 1][tid];
      int e = idx * 4;
      fn(e + 0, v.x); fn(e + 1, v.y); fn(e + 2, v.z); fn(e + 3, v.w);
    }
    __syncthreads();                        // protect buf before next overwrite
  }
}

__global__ __launch_bounds__(WG)
void vocab_project_topk_kernel(const float* __restrict__ logits,
                               const int*   __restrict__ mapping,
                               float*       __restrict__ out) {
  const int row = blockIdx.x;
  const int tid = threadIdx.x;
  const float4* row4 = (const float4*)(logits + (size_t)row * TEACHER_VOCAB);
  float* outp = out + (size_t)row * STUDENT_VOCAB;

  __shared__ float4   s_buf[2][TILEF4];   // 16 KB staging
  __shared__ unsigned s_hist[256];
  __shared__ unsigned s_maxkey, s_cnt, s_tie;
  __shared__ int      s_idx[TOPK];
  __shared__ int      s_sid[TOPK];
  __shared__ float    s_w[TOPK];
  __shared__ float    s_denom;

  // ---- Phase 0: zero this WG's output row ----
  float4 z; z.x = z.y = z.z = z.w = 0.0f;
  float4* out4 = (float4*)outp;
  for (int j = tid; j < STUDENT_VOCAB / 4; j += WG) out4[j] = z;
  __threadfence();   // zeros complete before the leader stores much later

  if (tid == 0) { s_maxkey = 0u; s_cnt = 0u; s_tie = 0u; }

  // ---- Phase A: 4-round radix select of the K-th largest key ----
  unsigned prefix = 0u, need = TOPK;
  for (int s = 24; s >= 0; s -= 8) {
    for (int j = tid; j < 256; j += WG) s_hist[j] = 0u;
    __syncthreads();
    unsigned lmax = 0u;
    stream_row(row4, s_buf, [&](int i, float v) {
      (void)i;
      unsigned k = f2k(v);
      if (s == 24) {
        if (k > lmax) lmax = k;
        atomicAdd(&s_hist[k >> 24], 1u);                     // ds_add_u32
      } else if ((k >> (s + 8)) == prefix) {
        atomicAdd(&s_hist[(k >> s) & 255u], 1u);
      }
    });
    if (s == 24) atomicMax(&s_maxkey, lmax);                 // ds_max_u32
    // Redundant per-thread descending scan (all threads get same answer).
    unsigned acc = 0u, g = 0u;
    for (int d = 255; d >= 0; --d) {
      unsigned h = s_hist[d];
      if (acc + h >= need) { g = (unsigned)d; break; }
      acc += h;
    }
    prefix = (prefix << 8) | g;
    need -= acc;          // invariant: need >= 1
    __syncthreads();      // scan done before next round clears hist
  }
  const unsigned thresh = prefix;   // exact 32-bit key of K-th largest element

  // ---- Phase B: collect exactly K entries (idx, student id, exp weight) ----
  const float ref = k2f(s_maxkey);  // exp reference: row max => no overflow
  stream_row(row4, s_buf, [&](int i, float v) {
    unsigned k = f2k(v);
    bool take;
    if (k > thresh)       take = true;
    else if (k == thresh) take = (atomicAdd(&s_tie, 1u) < need);
    else                  take = false;
    if (take) {
      unsigned pos = atomicAdd(&s_cnt, 1u);                  // ds_add_rtn_u32
      if (pos < TOPK) {
        int sid = mapping[i];
        sid = sid < 0 ? 0 : (sid >= STUDENT_VOCAB ? STUDENT_VOCAB - 1 : sid);
        s_idx[pos] = i;
        s_sid[pos] = sid;
        s_w[pos]   = __expf((v - ref) * INV_T);
      }
    }
  });

  // pad (defensive; s_cnt == TOPK by construction)
  unsigned nsel = s_cnt;
  if (nsel > TOPK) nsel = TOPK;
  for (int p = tid; p < TOPK; p += WG)
    if ((unsigned)p >= nsel) { s_idx[p] = 0x7FFFFFFF; s_sid[p] = 0; s_w[p] = 0.0f; }
  __syncthreads();

  // ---- Phase C: bitonic sort by teacher index => canonical deterministic order
  for (int ksz = 2; ksz <= TOPK; ksz <<= 1) {
    for (int j = ksz >> 1; j > 0; j >>= 1) {
      if (tid < TOPK) {
        int ixj = tid ^ j;
        if (ixj > tid) {
          int a = s_idx[tid], b = s_idx[ixj];
          bool up = ((tid & ksz) == 0);
          if (up ? (a > b) : (a < b)) {
            s_idx[tid] = b;            s_idx[ixj] = a;
            int   sa = s_sid[tid];     s_sid[tid] = s_sid[ixj]; s_sid[ixj] = sa;
            float wa = s_w[tid];       s_w[tid]   = s_w[ixj];   s_w[ixj]   = wa;
          }
        }
      }
      __syncthreads();
    }
  }

  // deterministic serial denominator
  if (tid == 0) {
    float sum = 0.0f;
    for (int p = 0; p < TOPK; ++p) sum += s_w[p];
    s_denom = sum < 1e-8f ? 1e-8f : sum;
  }
  __syncthreads();
  const float denom = s_denom;

  // ---- Phase D: leader-dedup per student id, plain (non-atomic) stores ----
  if (tid < TOPK) {
    const int sid = s_sid[tid];
    bool leader = true;
    for (int p = 0; p < tid; ++p)
      if (s_sid[p] == sid) { leader = false; break; }
    if (leader) {
      float a = s_w[tid];
      for (int p = tid + 1; p < TOPK; ++p)
        if (s_sid[p] == sid) a += s_w[p];
      outp[sid] = a / denom;
    }
  }
}

extern "C" void kernel_launch(void* const* d_in, const int* in_sizes, int n_in,
                              void* d_out, int out_size, void* d_ws, size_t ws_size,
                              hipStream_t stream) {
  (void)in_sizes; (void)n_in; (void)out_size; (void)d_ws; (void)ws_size;
  const float* logits  = (const float*)d_in[0];  // (2,1024,128000) f32
  const int*   mapping = (const int*)d_in[1];    // (128000,) int32
  float*       out     = (float*)d_out;          // (2,1024,32000) f32
  vocab_project_topk_kernel<<<dim3(NROWS), dim3(WG), 0, stream>>>(logits, mapping, out);
}